// RegionLoss_1494648619525
// MI455X (gfx1250) — compile-verified
//
#include <hip/hip_runtime.h>
#include <math.h>

// Problem constants (from reference)
#define KCORN 9
#define NB 64
#define NH 32
#define NW 32
#define NCELL (NH * NW)          // 1024
#define NCH 20                   // 2K+1+NC
#define NTGT 50
#define NLAB 21                  // 2K+3
#define TH_DIST 80.0f
#define SHARP 2.0f
#define IMW 640.0f
#define IMH 480.0f
#define SIL 0.6f
#define OBJW 5.0f
#define EPSV 1e-5f

typedef float v2f __attribute__((ext_vector_type(2)));
typedef float v8f __attribute__((ext_vector_type(8)));

__device__ __forceinline__ float sigmoidf_(float x) {
    return 1.0f / (1.0f + __expf(-x));
}

// ---------------------------------------------------------------------------
// Kernel 1: all-pairs corner-confidence via WMMA + NOOBJ conf loss.
// grid = NB*8 blocks, block = 256 threads (8 waves). Each block handles one
// batch b and 128 cells; each wave handles a 16-cell tile.
// dist^2(t,s) is produced directly by V_WMMA_F32_16X16X4_F32 with
//   A = (-2*gx, -2*gy, 0, 0), B = (px, py, 0, 0), C = |g|^2 + |p|^2.
// Loop structure: k rolled (unroll 1) to bound live C/D tiles at 4 (no
// spills); mt unrolled with a persistent acc[4][8] accumulator. All operand
// masking is done with selects on loaded values so EXEC stays all-ones.
// ---------------------------------------------------------------------------
__global__ void region_pairwise_kernel(const float* __restrict__ out,
                                       const float* __restrict__ tgt,
                                       const int* __restrict__ epoch,
                                       float* __restrict__ partial) {
    __shared__ float s_ax[64 * KCORN];   // gt_x * IMW  (scaled), padded to 64
    __shared__ float s_ay[64 * KCORN];   // gt_y * IMH
    __shared__ float s_gn[64 * KCORN];   // ax^2 + ay^2
    __shared__ int   s_valid[64];
    __shared__ int   s_tcell[64];        // flat target cell or -1
    __shared__ float s_red[256];

    const int b   = blockIdx.x >> 3;
    const int tid = threadIdx.x;

    // ---- setup: per-target data into LDS ----
    if (tid < 64) {
        const int t = tid;
        int   valid = 0;
        int   tc    = -1;
        if (t < NTGT) {
            const float* row = tgt + b * (NTGT * NLAB) + t * NLAB;
            // valid = cumprod(tgt[:, :, 1] != 0) > 0
            int v = 1;
            for (int u = 0; u <= t; ++u)
                v = v && (tgt[b * (NTGT * NLAB) + u * NLAB + 1] != 0.0f);
            valid = v;
            if (valid) {
                const float gx0 = row[1] * (float)NW;
                const float gy0 = row[2] * (float)NH;
                const int gi0 = (int)floorf(gx0);
                const int gj0 = (int)floorf(gy0);
                if (gi0 >= 0 && gi0 < NW && gj0 >= 0 && gj0 < NH)
                    tc = gj0 * NW + gi0;
            }
            for (int k = 0; k < KCORN; ++k) {
                float ax = valid ? row[1 + 2 * k] * IMW : 0.0f;
                float ay = valid ? row[2 + 2 * k] * IMH : 0.0f;
                s_ax[t * KCORN + k] = ax;
                s_ay[t * KCORN + k] = ay;
                s_gn[t * KCORN + k] = ax * ax + ay * ay;
            }
        } else {
            for (int k = 0; k < KCORN; ++k) {
                s_ax[t * KCORN + k] = 0.0f;
                s_ay[t * KCORN + k] = 0.0f;
                s_gn[t * KCORN + k] = 0.0f;
            }
        }
        s_valid[t] = valid;
        s_tcell[t] = tc;
    }
    __syncthreads();

    const int wave  = tid >> 5;
    const int lane  = tid & 31;
    const int ntile = (blockIdx.x & 7) * 8 + wave;    // 0..63
    const int sloc  = ntile * 16 + (lane & 15);       // this lane's cell
    const int gi    = sloc & (NW - 1);
    const int gj    = sloc >> 5;
    const bool lo   = (lane < 16);
    const int  hb   = lo ? 0 : 8;                     // C/D row base per half

    const float* ob = out + (size_t)b * NCH * NCELL;

    const float invD  = 1.0f / (__expf(SHARP) - 1.0f + EPSV);
    const float invTH = 1.0f / TH_DIST;

    float acc[4][8];
#pragma unroll
    for (int mt = 0; mt < 4; ++mt)
#pragma unroll
        for (int v = 0; v < 8; ++v) acc[mt][v] = 0.0f;

#pragma unroll 1
    for (int k = 0; k < KCORN; ++k) {
        // this lane's predicted corner k, scaled to pixels
        float x = ob[(2 * k) * NCELL + sloc];
        float y = ob[(2 * k + 1) * NCELL + sloc];
        if (k == 0) { x = sigmoidf_(x); y = sigmoidf_(y); }
        const float pxs = (x + (float)gi) * (IMW / (float)NW);   // *20
        const float pys = (y + (float)gj) * (IMH / (float)NH);   // *15
        const float pn  = pxs * pxs + pys * pys;

        v2f bb;
        bb.x = lo ? pxs : 0.0f;   // B rows K=0/K=1 in lanes 0-15, K=2/3 zero
        bb.y = lo ? pys : 0.0f;

#pragma unroll
        for (int mt = 0; mt < 4; ++mt) {
            const int tA  = mt * 16 + (lane & 15);
            // unconditional LDS loads, then select -> v_cndmask (EXEC stays full)
            const float axv = s_ax[tA * KCORN + k];
            const float ayv = s_ay[tA * KCORN + k];
            v2f a;
            a.x = lo ? -2.0f * axv : 0.0f;
            a.y = lo ? -2.0f * ayv : 0.0f;

            v8f c;
            const int tb = mt * 16 + hb;
#pragma unroll
            for (int v = 0; v < 8; ++v)
                c[v] = s_gn[(tb + v) * KCORN + k] + pn;        // |g|^2 + |p|^2

            // D = A*B + C  ->  dist^2 per (target, cell)
            c = __builtin_amdgcn_wmma_f32_16x16x4_f32(
                    false, a, false, bb, (short)0, c, false, false);

#pragma unroll
            for (int v = 0; v < 8; ++v) {
                const float d2   = fmaxf(c[v], 0.0f);
                const float dist = __builtin_sqrtf(d2);
                const float cf   = (__expf(SHARP * (1.0f - dist * invTH)) - 1.0f) * invD;
                acc[mt][v] += (dist < TH_DIST) ? cf : 0.0f;
            }
        }
    }

    float curmax = 0.0f;
#pragma unroll
    for (int mt = 0; mt < 4; ++mt) {
#pragma unroll
        for (int v = 0; v < 8; ++v) {
            const int tv = mt * 16 + v + hb;
            const float m = acc[mt][v] * (1.0f / (float)KCORN);
            curmax = fmaxf(curmax, s_valid[tv] ? m : 0.0f);
        }
    }

    // combine the two half-wave target partitions (same cell)
    const float other = __shfl_xor(curmax, 16, 32);
    const float cur   = fmaxf(curmax, other);

    // NOOBJ conf loss for this cell (lanes 0-15 own distinct cells)
    float contrib = 0.0f;
    {
        const float confc = sigmoidf_(ob[(2 * KCORN) * NCELL + sloc]);
        bool ist = false;
        for (int t = 0; t < 64; ++t)
            ist = ist || (s_valid[t] && s_tcell[t] == sloc);
        const bool take = lo && !ist && (cur <= SIL);
        contrib = take ? 0.5f * confc * confc : 0.0f;
    }
    if (*epoch <= 15) contrib = 0.0f;

    // block reduction -> one partial per block
    s_red[tid] = contrib;
    __syncthreads();
    for (int off = 128; off > 0; off >>= 1) {
        if (tid < off) s_red[tid] += s_red[tid + off];
        __syncthreads();
    }
    if (tid == 0) partial[blockIdx.x] = s_red[0];
}

// ---------------------------------------------------------------------------
// Kernel 2: per-target coord loss + OBJ conf loss. grid = NB, block = 64.
// ---------------------------------------------------------------------------
__global__ void region_target_kernel(const float* __restrict__ out,
                                     const float* __restrict__ tgt,
                                     const int* __restrict__ epoch,
                                     float* __restrict__ partial) {
    __shared__ float s_red[64];
    const int b = blockIdx.x;
    const int t = threadIdx.x;

    float contrib = 0.0f;
    if (t < NTGT) {
        const float* row = tgt + b * (NTGT * NLAB) + t * NLAB;
        int valid = 1;
        for (int u = 0; u <= t; ++u)
            valid = valid && (tgt[b * (NTGT * NLAB) + u * NLAB + 1] != 0.0f);
        if (valid) {
            const float gx0 = row[1] * (float)NW;
            const float gy0 = row[2] * (float)NH;
            const int gi0 = (int)floorf(gx0);
            const int gj0 = (int)floorf(gy0);
            if (gi0 >= 0 && gi0 < NW && gj0 >= 0 && gj0 < NH) {
                const int s = gj0 * NW + gi0;
                const float* ob = out + (size_t)b * NCH * NCELL;
                const float invD  = 1.0f / (__expf(SHARP) - 1.0f + EPSV);
                const float invTH = 1.0f / TH_DIST;
                float lxy = 0.0f, cg = 0.0f;
#pragma unroll
                for (int k = 0; k < KCORN; ++k) {
                    float x = ob[(2 * k) * NCELL + s];
                    float y = ob[(2 * k + 1) * NCELL + s];
                    if (k == 0) { x = sigmoidf_(x); y = sigmoidf_(y); }
                    const float gxk = row[1 + 2 * k] * (float)NW;
                    const float gyk = row[2 + 2 * k] * (float)NH;
                    const float txk = gxk - (float)gi0;
                    const float tyk = gyk - (float)gj0;
                    const float dx = x - txk, dy = y - tyk;
                    lxy += dx * dx + dy * dy;
                    // conf_gt: gt vs predicted corner at this cell
                    const float pxn = (x + (float)gi0) * (1.0f / (float)NW);
                    const float pyn = (y + (float)gj0) * (1.0f / (float)NH);
                    const float ddx = (row[1 + 2 * k] - pxn) * IMW;
                    const float ddy = (row[2 + 2 * k] - pyn) * IMH;
                    const float dist = __builtin_sqrtf(ddx * ddx + ddy * ddy);
                    const float cf = (__expf(SHARP * (1.0f - dist * invTH)) - 1.0f) * invD;
                    cg += (dist < TH_DIST) ? cf : 0.0f;
                }
                cg *= (1.0f / (float)KCORN);
                const float confc = sigmoidf_(ob[(2 * KCORN) * NCELL + s]);
                contrib = 0.5f * lxy;
                if (*epoch > 15) {
                    const float dcf = confc - cg;
                    contrib += 0.5f * OBJW * dcf * dcf;
                }
            }
        }
    }

    s_red[t] = contrib;
    __syncthreads();
    for (int off = 32; off > 0; off >>= 1) {
        if (t < off) s_red[t] += s_red[t + off];
        __syncthreads();
    }
    if (t == 0) partial[NB * 8 + b] = s_red[0];
}

// ---------------------------------------------------------------------------
// Kernel 3: deterministic fixed-order final sum of all partials.
// ---------------------------------------------------------------------------
__global__ void region_final_kernel(const float* __restrict__ partial,
                                    float* __restrict__ d_out) {
    if (threadIdx.x == 0 && blockIdx.x == 0) {
        float sum = 0.0f;
        for (int i = 0; i < NB * 8 + NB; ++i) sum += partial[i];
        d_out[0] = sum;
    }
}

extern "C" void kernel_launch(void* const* d_in, const int* in_sizes, int n_in,
                              void* d_out, int out_size, void* d_ws, size_t ws_size,
                              hipStream_t stream) {
    (void)in_sizes; (void)n_in; (void)out_size; (void)ws_size;
    const float* out_t = (const float*)d_in[0];   // (64, 20, 32, 32) fp32
    const float* tgt   = (const float*)d_in[1];   // (64, 1050) fp32
    const int*   epoch = (const int*)d_in[2];     // scalar
    float* loss    = (float*)d_out;
    float* partial = (float*)d_ws;                // 512 + 64 floats

    region_pairwise_kernel<<<dim3(NB * 8), dim3(256), 0, stream>>>(out_t, tgt, epoch, partial);
    region_target_kernel<<<dim3(NB), dim3(64), 0, stream>>>(out_t, tgt, epoch, partial);
    region_final_kernel<<<dim3(1), dim3(32), 0, stream>>>(partial, loss);
}